// BallePdfEstim_19567871001061
// MI455X (gfx1250) — compile-verified
//
#include <hip/hip_runtime.h>
#include <hip/hip_bf16.h>
#include <math.h>

// ---------------------------------------------------------------------------
// Types
// ---------------------------------------------------------------------------
typedef float v2f __attribute__((ext_vector_type(2)));
typedef float v4f __attribute__((ext_vector_type(4)));

#define NB 16
#define NC 192
#define HW 16384            // 128*128
#define CHUNK 4096          // elements per block (256 thr * 16 elem)
#define PSTRIDE 48          // padded per-channel param stride (floats)

// ---------------------------------------------------------------------------
// Fast transcendentals (gfx1250): v_tanh_f32 / v_exp_f32 / v_rcp_f32
// Device-pass-only probes: a #warning below in stderr means the fallback
// path is active on the DEVICE compile (host pass is excluded via __AMDGCN__).
// ---------------------------------------------------------------------------
#if __has_builtin(__builtin_amdgcn_rcpf)
#define RCPF(x) __builtin_amdgcn_rcpf(x)
#else
#if defined(__AMDGCN__)
#warning "CDNA5 probe: __builtin_amdgcn_rcpf NOT available on device; using 1/x"
#endif
#define RCPF(x) (1.0f / (x))
#endif

// HIP always provides the fast native exp intrinsic on device
#define FEXP(x) __expf(x)

#if __has_builtin(__builtin_amdgcn_tanhf)
#define FTANH(x) __builtin_amdgcn_tanhf(x)
#else
#if defined(__AMDGCN__)
#warning "CDNA5 probe: __builtin_amdgcn_tanhf NOT available on device; using exp-based tanh"
#endif
// tanh(x) = 1 - 2/(exp(2x) + 1)
#define FTANH(x) (1.0f - 2.0f * RCPF(FEXP(2.0f * (x)) + 1.0f))
#endif

__device__ __forceinline__ v2f tanh2(v2f t) {
  v2f r;
  r.x = FTANH(t.x);
  r.y = FTANH(t.y);
  return r;
}

__device__ __forceinline__ v2f sigmoid2(v2f t) {
  v2f r;
  r.x = RCPF(1.0f + FEXP(-t.x));
  r.y = RCPF(1.0f + FEXP(-t.y));
  return r;
}

// ---------------------------------------------------------------------------
// HOT kernel (placed first so the disasm snippet shows it).
// One block = 4096 contiguous elements of one (b,c) plane.
// x staged via gfx1250 async global->LDS B128 copies (ASYNCcnt pipeline),
// params via uniform (SMEM) loads into SGPRs, both CDF branches packed in v2f
// so the backend can emit v_pk_fma_f32. Output via non-temporal B128 stores.
// ---------------------------------------------------------------------------
__global__ __launch_bounds__(256) void balle_main_kernel(const float* __restrict__ x,
                                                         const float* __restrict__ P,
                                                         float* __restrict__ out) {
  __shared__ v4f tile[4][256];

  const int bid = blockIdx.x;
  const int plane = bid >> 2;        // b*NC + c
  const int chunk = bid & 3;
  const int ch = plane % NC;
  const unsigned tid = threadIdx.x;

  const float* __restrict__ xbase = x + (size_t)plane * HW + (size_t)chunk * CHUNK;
  float* __restrict__ obase = out + (size_t)plane * HW + (size_t)chunk * CHUNK;

  // ---- issue 4 async B128 global->LDS copies (each lane stages its own 16B)
#pragma unroll
  for (int i = 0; i < 4; ++i) {
    unsigned lds = (unsigned)(uintptr_t)(&tile[i][tid]);
    unsigned voff = (unsigned)i * 4096u + tid * 16u;  // bytes
    asm volatile("global_load_async_to_lds_b128 %0, %1, %2"
                 :
                 : "v"(lds), "v"(voff), "s"(xbase)
                 : "memory");
  }

  // ---- block-uniform channel params -> scalar (SGPR) loads
  const float* __restrict__ p = P + ch * PSTRIDE;
  const float H00 = p[0], H01 = p[1], H02 = p[2];
  const float B00 = p[3], B01 = p[4], B02 = p[5];
  const float A00 = p[6], A01 = p[7], A02 = p[8];
  float W1[9], W2[9];
#pragma unroll
  for (int j = 0; j < 9; ++j) {
    W1[j] = p[9 + j];
    W2[j] = p[24 + j];
  }
  const float B10 = p[18], B11 = p[19], B12 = p[20];
  const float A10 = p[21], A11 = p[22], A12 = p[23];
  const float B20 = p[33], B21 = p[34], B22 = p[35];
  const float A20 = p[36], A21 = p[37], A22 = p[38];
  const float H30 = p[39], H31 = p[40], H32 = p[41];
  const float B3v = p[42];

  // both CDF branches (x+0.5 / x-0.5) packed per v2f lane pair
  auto eval = [&](float xv) -> float {
    v2f v;
    v.x = xv + 0.5f;
    v.y = xv - 0.5f;
    // layer 0: 1 -> 3
    v2f t0 = v * H00 + B00;
    v2f t1 = v * H01 + B01;
    v2f t2 = v * H02 + B02;
    v2f u0 = t0 + A00 * tanh2(t0);
    v2f u1 = t1 + A01 * tanh2(t1);
    v2f u2 = t2 + A02 * tanh2(t2);
    // layer 1: 3 -> 3   (h[d][r], row-major d*3+r)
    v2f s0 = u0 * W1[0] + u1 * W1[3] + u2 * W1[6] + B10;
    v2f s1 = u0 * W1[1] + u1 * W1[4] + u2 * W1[7] + B11;
    v2f s2 = u0 * W1[2] + u1 * W1[5] + u2 * W1[8] + B12;
    v2f q0 = s0 + A10 * tanh2(s0);
    v2f q1 = s1 + A11 * tanh2(s1);
    v2f q2 = s2 + A12 * tanh2(s2);
    // layer 2: 3 -> 3
    v2f r0 = q0 * W2[0] + q1 * W2[3] + q2 * W2[6] + B20;
    v2f r1 = q0 * W2[1] + q1 * W2[4] + q2 * W2[7] + B21;
    v2f r2 = q0 * W2[2] + q1 * W2[5] + q2 * W2[8] + B22;
    v2f g0 = r0 + A20 * tanh2(r0);
    v2f g1 = r1 + A21 * tanh2(r1);
    v2f g2 = r2 + A22 * tanh2(r2);
    // layer 3: 3 -> 1, sigmoid, difference of the two packed branches
    v2f y = g0 * H30 + g1 * H31 + g2 * H32 + B3v;
    v2f sg = sigmoid2(y);
    return sg.x - sg.y;
  };

#pragma unroll
  for (int i = 0; i < 4; ++i) {
    // staggered waits: compute on tile i while tiles i+1..3 are in flight
    if (i == 0)
      asm volatile("s_wait_asynccnt 0x3" ::: "memory");
    else if (i == 1)
      asm volatile("s_wait_asynccnt 0x2" ::: "memory");
    else if (i == 2)
      asm volatile("s_wait_asynccnt 0x1" ::: "memory");
    else
      asm volatile("s_wait_asynccnt 0x0" ::: "memory");

    v4f xin = tile[i][tid];  // ds_load_b128, own slot -> no barrier needed
    v4f r;
    r.x = eval(xin.x);
    r.y = eval(xin.y);
    r.z = eval(xin.z);
    r.w = eval(xin.w);
    __builtin_nontemporal_store(r, (v4f*)obase + (size_t)i * 256 + tid);
  }
}

// ---------------------------------------------------------------------------
// Kernel 2: per-channel parameter prep (softplus of weights, tanh of gates).
// Runs once over 192 channels -> negligible runtime; precise math is fine.
// Layout per channel (PSTRIDE floats):
//  [0:3)=sp(h0) [3:6)=b0 [6:9)=tanh(a0) [9:18)=sp(h1) [18:21)=b1 [21:24)=tanh(a1)
//  [24:33)=sp(h2) [33:36)=b2 [36:39)=tanh(a2) [39:42)=sp(h3) [42]=b3
// ---------------------------------------------------------------------------
__device__ __forceinline__ float softplus_(float v) {
  // numerically stable softplus
  return (v > 20.0f) ? v : log1pf(expf(v));
}

__global__ void balle_prep_kernel(const float* __restrict__ h0, const float* __restrict__ h1,
                                  const float* __restrict__ h2, const float* __restrict__ h3,
                                  const float* __restrict__ a0, const float* __restrict__ a1,
                                  const float* __restrict__ a2, const float* __restrict__ b0,
                                  const float* __restrict__ b1, const float* __restrict__ b2,
                                  const float* __restrict__ b3, float* __restrict__ P) {
  int c = blockIdx.x * blockDim.x + threadIdx.x;
  if (c >= NC) return;
  float* p = P + c * PSTRIDE;
#pragma unroll
  for (int j = 0; j < 3; ++j) p[0 + j] = softplus_(h0[c * 3 + j]);
#pragma unroll
  for (int j = 0; j < 3; ++j) p[3 + j] = b0[c * 3 + j];
#pragma unroll
  for (int j = 0; j < 3; ++j) p[6 + j] = tanhf(a0[c * 3 + j]);
#pragma unroll
  for (int j = 0; j < 9; ++j) p[9 + j] = softplus_(h1[c * 9 + j]);
#pragma unroll
  for (int j = 0; j < 3; ++j) p[18 + j] = b1[c * 3 + j];
#pragma unroll
  for (int j = 0; j < 3; ++j) p[21 + j] = tanhf(a1[c * 3 + j]);
#pragma unroll
  for (int j = 0; j < 9; ++j) p[24 + j] = softplus_(h2[c * 9 + j]);
#pragma unroll
  for (int j = 0; j < 3; ++j) p[33 + j] = b2[c * 3 + j];
#pragma unroll
  for (int j = 0; j < 3; ++j) p[36 + j] = tanhf(a2[c * 3 + j]);
#pragma unroll
  for (int j = 0; j < 3; ++j) p[39 + j] = softplus_(h3[c * 3 + j]);
  p[42] = b3[c];
  p[43] = p[44] = p[45] = p[46] = p[47] = 0.0f;
}

// ---------------------------------------------------------------------------
// Launch
// ---------------------------------------------------------------------------
extern "C" void kernel_launch(void* const* d_in, const int* in_sizes, int n_in,
                              void* d_out, int out_size, void* d_ws, size_t ws_size,
                              hipStream_t stream) {
  (void)in_sizes;
  (void)n_in;
  (void)out_size;
  (void)ws_size;

  const float* x  = (const float*)d_in[0];
  const float* h0 = (const float*)d_in[1];
  const float* h1 = (const float*)d_in[2];
  const float* h2 = (const float*)d_in[3];
  const float* h3 = (const float*)d_in[4];
  const float* a0 = (const float*)d_in[5];
  const float* a1 = (const float*)d_in[6];
  const float* a2 = (const float*)d_in[7];
  const float* b0 = (const float*)d_in[8];
  const float* b1 = (const float*)d_in[9];
  const float* b2 = (const float*)d_in[10];
  const float* b3 = (const float*)d_in[11];

  float* P = (float*)d_ws;  // NC * PSTRIDE floats = 36,864 B

  balle_prep_kernel<<<1, NC, 0, stream>>>(h0, h1, h2, h3, a0, a1, a2, b0, b1, b2, b3, P);

  const int nblocks = NB * NC * (HW / CHUNK);  // 12288
  balle_main_kernel<<<nblocks, 256, 0, stream>>>(x, P, (float*)d_out);
}